// UsmFilter_58737972740581
// MI455X (gfx1250) — compile-verified
//
#include <hip/hip_runtime.h>
#include <math.h>

typedef float v2f __attribute__((ext_vector_type(2)));
typedef float v8f __attribute__((ext_vector_type(8)));

#define RADIUS  12
#define KSIZE   25
#define TILE    64
#define PAD_DIM 88              // TILE + 2*RADIUS
#define SROWS   96              // padded-row tiles: 6 x 16
#define SP      89              // odd stride -> no LDS bank conflicts
#define HROWS   96
#define HP      65              // odd stride
#define IMG_W   512
#define PLANE   (IMG_W * IMG_W)
#define WEXT_OFF 16             // Wext[i+WEXT_OFF] = w[i], zero outside [0,25)

__device__ __forceinline__ int reflect512(int i) {
    i = (i < 0) ? -i : i;
    i = (i > 511) ? (1022 - i) : i;
    return i;
}

__global__ __launch_bounds__(256)
void usm_wmma_kernel(const float* __restrict__ img,
                     const float* __restrict__ feat,
                     float* __restrict__ out) {
    __shared__ float S[SROWS * SP];   // padded input tile (fp32)
    __shared__ float Hs[HROWS * HP];  // horizontal-pass intermediate
    __shared__ float Wext[64];        // zero-padded 1D Gaussian, index i+16

    const int tid = threadIdx.x;
    const int blk = blockIdx.x;
    const int tx0 = (blk & 7) * TILE;
    const int ty0 = ((blk >> 3) & 7) * TILE;
    const int ch  = (blk >> 6) % 3;
    const int b   = blk / 192;

    const float* plane = img + (size_t)(b * 3 + ch) * PLANE;

    // ---- zero-padded normalized 1D Gaussian (radius 12, sigma 5) ----
    if (tid == 0) {
        float tmp[KSIZE];
        float s = 0.f;
        #pragma unroll
        for (int i = 0; i < KSIZE; ++i) {
            float x = (float)(i - RADIUS) * (1.0f / 5.0f);
            tmp[i] = __expf(-0.5f * x * x);
            s += tmp[i];
        }
        float inv = 1.0f / s;
        #pragma unroll
        for (int i = 0; i < 64; ++i) {
            int k = i - WEXT_OFF;
            Wext[i] = (k >= 0 && k < KSIZE) ? tmp[k] * inv : 0.0f;
        }
    }

    // ---- stage reflect-padded 88x88 tile into LDS ----
    for (int i = tid; i < PAD_DIM * PAD_DIM; i += 256) {
        int r = i / PAD_DIM;
        int c = i - r * PAD_DIM;
        int gy = reflect512(ty0 - RADIUS + r);
        int gx = reflect512(tx0 - RADIUS + c);
        S[r * SP + c] = plane[gy * IMG_W + gx];
    }
    // zero the 8 tail rows (read by the last 16-row WMMA strip, results unused)
    for (int i = tid; i < (SROWS - PAD_DIM) * SP; i += 256)
        S[PAD_DIM * SP + i] = 0.0f;

    __syncthreads();

    const int lane = tid & 31;
    const int ln   = lane & 15;       // row (A) / column (B,C,D) within tile
    const int hi   = lane >> 4;       // half-wave K-pair selector
    const int wv   = tid >> 5;        // wave id 0..7

    // ---- hoist banded-weight operands into registers (per-lane constant).
    // Same values serve as B-matrix (horizontal) and A-matrix (vertical):
    // band[k][ln] = w[k - ln], zero-padded. Unconditional LDS loads.
    float wb[20];
    #pragma unroll
    for (int kc = 0; kc < 10; ++kc) {
        int kb = kc * 4 + hi * 2;
        wb[2 * kc]     = Wext[WEXT_OFF + kb - ln];
        wb[2 * kc + 1] = Wext[WEXT_OFF + kb + 1 - ln];
    }

    // ---- horizontal pass: H(16x16) = I(16x40) x Wband(40x16), K = 10x4 ----
    // 6 row-tiles x 4 col-tiles = 24 tiles; fixed 3 per wave (no divergence)
    #pragma unroll
    for (int tt = 0; tt < 3; ++tt) {
        int t  = wv + tt * 8;
        int rb = (t >> 2) * 16;
        int cb = (t & 3) * 16;
        v8f acc = {};
        #pragma unroll
        for (int kc = 0; kc < 10; ++kc) {
            int kb = kc * 4 + hi * 2;
            const float* sp = &S[(rb + ln) * SP + cb + kb];
            v2f a;  a.x  = sp[0];          a.y  = sp[1];
            v2f bb; bb.x = wb[2 * kc];     bb.y = wb[2 * kc + 1];
            acc = __builtin_amdgcn_wmma_f32_16x16x4_f32(
                false, a, false, bb, (short)0, acc, false, false);
        }
        #pragma unroll
        for (int v = 0; v < 8; ++v)        // D: row v+8*hi, col ln
            Hs[(rb + v + 8 * hi) * HP + cb + ln] = acc[v];
    }

    __syncthreads();

    float pf    = feat[b * 8];
    float param = (tanhf(pf) * 0.5f + 0.5f) * 5.0f;

    // ---- vertical pass: B(16x16) = Wband(16x40) x H(40x16) + epilogue ----
    // 4 x 4 = 16 tiles; fixed 2 per wave
    #pragma unroll
    for (int tt = 0; tt < 2; ++tt) {
        int t  = wv + tt * 8;
        int rb = (t >> 2) * 16;
        int cb = (t & 3) * 16;
        v8f acc = {};
        #pragma unroll
        for (int kc = 0; kc < 10; ++kc) {
            int kb = kc * 4 + hi * 2;
            v2f a;  a.x  = wb[2 * kc];     a.y  = wb[2 * kc + 1];
            const float* hp = &Hs[(rb + kb) * HP + cb + ln];
            v2f bb; bb.x = hp[0];          bb.y = hp[HP];
            acc = __builtin_amdgcn_wmma_f32_16x16x4_f32(
                false, a, false, bb, (short)0, acc, false, false);
        }
        #pragma unroll
        for (int v = 0; v < 8; ++v) {
            int r = rb + v + 8 * hi;
            int c = cb + ln;
            float orig = S[(r + RADIUS) * SP + c + RADIUS];
            float val  = orig + param * (orig - acc[v]);
            out[(size_t)(b * 3 + ch) * PLANE + (size_t)(ty0 + r) * IMG_W + (tx0 + c)] = val;
        }
    }
}

__global__ void usm_param_kernel(const float* __restrict__ feat,
                                 float* __restrict__ out) {
    int b = threadIdx.x;
    if (b < 16) {
        float pf = feat[b * 8];
        out[(size_t)16 * 3 * PLANE + b] = (tanhf(pf) * 0.5f + 0.5f) * 5.0f;
    }
}

extern "C" void kernel_launch(void* const* d_in, const int* in_sizes, int n_in,
                              void* d_out, int out_size, void* d_ws, size_t ws_size,
                              hipStream_t stream) {
    (void)in_sizes; (void)n_in; (void)d_ws; (void)ws_size; (void)out_size;
    const float* img  = (const float*)d_in[0];
    const float* feat = (const float*)d_in[1];
    float* out = (float*)d_out;

    // 16 batches * 3 channels * (512/64)^2 tiles = 3072 blocks, 8 wave32s each
    usm_wmma_kernel<<<3072, 256, 0, stream>>>(img, feat, out);
    usm_param_kernel<<<1, 16, 0, stream>>>(feat, out);
}